// Baseline_635655160228
// MI455X (gfx1250) — compile-verified
//
#include <hip/hip_runtime.h>

// Problem constants (match reference)
#define B_     64
#define WAYS_  10
#define SHOTS_ 5
#define C_     128
#define HW_    1024
#define NTOT_  (SHOTS_ * HW_)   // 5120

typedef __attribute__((ext_vector_type(2))) float v2f;
typedef __attribute__((ext_vector_type(8))) float v8f;

// TDM descriptor argument types (probe-verified arity; see CDNA5_HIP.md)
typedef __attribute__((ext_vector_type(4))) unsigned int u32x4;
typedef __attribute__((ext_vector_type(8))) int          i32x8;
typedef __attribute__((ext_vector_type(4))) int          i32x4;

__device__ __forceinline__ v8f zero_v8f() {
    v8f z = {0.f, 0.f, 0.f, 0.f, 0.f, 0.f, 0.f, 0.f};
    return z;
}

// D = A(16x4, f32) * B(4x16, f32) + C(16x16, f32)
// 8-arg form: (neg_a, A, neg_b, B, c_mod, C, reuse_a, reuse_b)
__device__ __forceinline__ v8f wmma_f32_4(v2f a, v2f b, v8f c) {
    return __builtin_amdgcn_wmma_f32_16x16x4_f32(
        /*neg_a=*/false, a, /*neg_b=*/false, b,
        /*c_mod=*/(short)0, c, /*reuse_a=*/false, /*reuse_b=*/false);
}

// ---------------------------------------------------------------------------
// TDM: DMA a 128x128 f32 tile (row-major, contiguous) from global into LDS.
// Builds the D# per CDNA5 ISA:
//   group0: count=1 | lds_addr | global_addr(57b) | type=2
//   group1: data_size=4B, tensor 128x128, tile 128x128, dim0_stride=128
//   groups 2/3: zero (tensor <= 2D)
// Issued by ONE wave (EXEC all ones); completion via s_wait_tensorcnt then
// a workgroup barrier publishes the LDS tile to the other waves.
// ---------------------------------------------------------------------------
__device__ __forceinline__ void tdm_load_cov_tile(const float* gsrc, float* lds_dst) {
    const unsigned long long gaddr = (unsigned long long)(uintptr_t)gsrc;
    // generic LDS pointer: low 32 bits are the LDS byte offset
    const unsigned lds_off = (unsigned)(uintptr_t)(void*)lds_dst;

    u32x4 g0;
    g0[0] = 1u;                                           // count=1 (valid), user mode
    g0[1] = lds_off;                                      // lds_addr (bytes)
    g0[2] = (unsigned)(gaddr & 0xFFFFFFFFull);            // global_addr[31:0]
    g0[3] = (unsigned)((gaddr >> 32) & 0x01FFFFFFull)     // global_addr[56:32]
          | 0x80000000u;                                  // type=2 ("image") at bits 127:126

    i32x8 g1;
    g1[0] = (int)(2u << 16);        // workgroup_mask=0, data_size=2 (4 bytes)
    g1[1] = (int)(128u << 16);      // tensor_dim0[15:0]=128 (atomic_barrier_addr=0)
    g1[2] = (int)(128u << 16);      // tensor_dim0[31:16]=0 | tensor_dim1[15:0]=128
    g1[3] = (int)(128u << 16);      // tensor_dim1[31:16]=0 | tile_dim0=128
    g1[4] = (int)128;               // tile_dim1=128 | tile_dim2=0
    g1[5] = (int)128;               // tensor_dim0_stride[31:0]=128 elements
    g1[6] = 0;                      // stride0 hi | tensor_dim1_stride lo
    g1[7] = 0;                      // tensor_dim1_stride hi

    i32x4 g2; g2[0] = 0; g2[1] = 0; g2[2] = 0; g2[3] = 0;
    i32x4 g3; g3[0] = 0; g3[1] = 0; g3[2] = 0; g3[3] = 0;

#if defined(__clang_major__) && (__clang_major__ >= 23)
    i32x8 g4;
    g4[0] = 0; g4[1] = 0; g4[2] = 0; g4[3] = 0;
    g4[4] = 0; g4[5] = 0; g4[6] = 0; g4[7] = 0;
    __builtin_amdgcn_tensor_load_to_lds(g0, g1, g2, g3, g4, 0);   // 6-arg (therock)
#else
    __builtin_amdgcn_tensor_load_to_lds(g0, g1, g2, g3, 0);       // 5-arg (ROCm 7.2)
#endif
    __builtin_amdgcn_s_wait_tensorcnt(0);
}

// ---------------------------------------------------------------------------
// Kernel 0: zero the output accumulator (harness poisons d_out with 0xAA)
// ---------------------------------------------------------------------------
__global__ void zero_kernel(float* __restrict__ out, int n) {
    int i = blockIdx.x * blockDim.x + threadIdx.x;
    if (i < n) out[i] = 0.f;
}

// ---------------------------------------------------------------------------
// Kernel 1: L2-normalize each (b,c) row of q over its 1024 pixels -> qn
// one 256-thread block per row; each thread handles 4 contiguous floats
// ---------------------------------------------------------------------------
__global__ void rownorm_kernel(const float* __restrict__ q, float* __restrict__ qn) {
    __shared__ float red[256];
    const int row = blockIdx.x;                 // b*C + c
    const float* src = q  + (size_t)row * HW_;
    float*       dst = qn + (size_t)row * HW_;
    const int t = threadIdx.x;

    float4 v = ((const float4*)src)[t];
    red[t] = v.x * v.x + v.y * v.y + v.z * v.z + v.w * v.w;
    __syncthreads();
    for (int s = 128; s > 0; s >>= 1) {
        if (t < s) red[t] += red[t + s];
        __syncthreads();
    }
    const float rn = 1.0f / sqrtf(red[0]);
    float4 o;
    o.x = v.x * rn; o.y = v.y * rn; o.z = v.z * rn; o.w = v.w * rn;
    ((float4*)dst)[t] = o;
}

// ---------------------------------------------------------------------------
// Kernel 2: per-class channel covariance via f32 WMMA Gram GEMM
// grid = WAYS blocks, 256 threads = 8 waves.
// cov[j] = (S~ S~^T)/(N-1), S~ = centered [128 x 5120], chunked by 64 in LDS.
// Wave w owns output row-tile mt = w (16 rows), loops all 8 column tiles.
// ---------------------------------------------------------------------------
__global__ void cov_kernel(const float* __restrict__ sup, float* __restrict__ cov) {
    __shared__ float meanv[C_];
    __shared__ float part[256];
    __shared__ float S[C_ * 64];          // centered chunk, 32 KB

    const int j = blockIdx.x;
    const int t = threadIdx.x;

    // --- per-channel mean over N = 5120 (2 threads per channel) ---
    {
        const int c = t >> 1, hh = t & 1;
        float s = 0.f;
        for (int shot = 0; shot < SHOTS_; ++shot) {
            const float* rp = sup + (((size_t)j * SHOTS_ + shot) * C_ + c) * HW_ + hh * 512;
            for (int pp = 0; pp < 512; ++pp) s += rp[pp];
        }
        part[t] = s;
    }
    __syncthreads();
    if (t < C_) meanv[t] = (part[2 * t] + part[2 * t + 1]) * (1.0f / (float)NTOT_);

    const int wave = t >> 5, lane = t & 31, ln = lane & 15, hf = lane >> 4;

    v8f acc[8];
#pragma unroll
    for (int nt = 0; nt < 8; ++nt) acc[nt] = zero_v8f();

    for (int n0 = 0; n0 < NTOT_; n0 += 64) {
        __syncthreads();   // guards meanv (1st iter) and S reuse (later iters)
        // load centered chunk S[c][kk] (coalesced: consecutive threads -> consecutive kk)
        for (int e = t; e < C_ * 64; e += 256) {
            const int c = e >> 6, kk = e & 63;
            const int n = n0 + kk;
            const int shot = n >> 10, pp = n & (HW_ - 1);
            S[e] = sup[(((size_t)j * SHOTS_ + shot) * C_ + c) * HW_ + pp] - meanv[c];
        }
        __syncthreads();

        for (int k0 = 0; k0 < 64; k0 += 4) {
            const int ka = k0 + 2 * hf;            // lanes 0-15: K=k0,k0+1; 16-31: k0+2,k0+3
            v2f a;
            a.x = S[(wave * 16 + ln) * 64 + ka];
            a.y = S[(wave * 16 + ln) * 64 + ka + 1];
#pragma unroll
            for (int nt = 0; nt < 8; ++nt) {
                v2f bb;                            // B[k][n] = S[ntile*16+n][k]
                bb.x = S[(nt * 16 + ln) * 64 + ka];
                bb.y = S[(nt * 16 + ln) * 64 + ka + 1];
                acc[nt] = wmma_f32_4(a, bb, acc[nt]);
            }
        }
    }

    // write cov: C/D layout -> VGPR i: lanes 0-15 M=i, lanes 16-31 M=8+i; N = lane%16
    const float inv = 1.0f / (float)(NTOT_ - 1);
    float* cj = cov + (size_t)j * C_ * C_;
#pragma unroll
    for (int nt = 0; nt < 8; ++nt) {
#pragma unroll
        for (int i = 0; i < 8; ++i) {
            cj[(wave * 16 + i + 8 * hf) * C_ + nt * 16 + ln] = acc[nt][i] * inv;
        }
    }
}

// ---------------------------------------------------------------------------
// Kernel 3: sim[b,j,p] = qn[:,p]^T cov[j] qn[:,p]  -> LeakyReLU -> dot conv_w
// grid = (8 pixel-tiles of 128, WAYS, B); 256 threads = 8 waves, 16 pixels/wave.
// cov[j] staged into 64 KB dynamic LDS by the Tensor Data Mover (wave 0);
// qn is L2-resident (33.5 MB << 192 MB L2) and prefetched.
// ---------------------------------------------------------------------------
__global__ void sim_kernel(const float* __restrict__ qn, const float* __restrict__ cov,
                           const float* __restrict__ cw, float* __restrict__ out) {
    extern __shared__ float cov_lds[];            // 128*128 f32 = 64 KB
    const int j = blockIdx.y, b = blockIdx.z;
    const int t = threadIdx.x;

    const float* cj = cov + (size_t)j * C_ * C_;
    const float* qb = qn + (size_t)b * C_ * HW_;  // [128][1024]

    // wave 0: async-tensor DMA of cov[j] into LDS (TENSORcnt-tracked)
    if (t < 32) {
        tdm_load_cov_tile(cj, cov_lds);
    }
    // meanwhile: prefetch this block's qn pixel columns (global_prefetch_b8)
    for (int r = t; r < C_; r += 256) {
        __builtin_prefetch((const void*)(qb + (size_t)r * HW_ + blockIdx.x * 128), 0, 0);
    }
    __syncthreads();                               // publish LDS tile to all waves

    const int wave = t >> 5, lane = t & 31, ln = lane & 15, hf = lane >> 4;
    const int p = blockIdx.x * 128 + wave * 16 + ln;   // this lane's pixel

    float partial = 0.f;
    for (int mt = 0; mt < 8; ++mt) {
        v8f acc = zero_v8f();
#pragma unroll 4
        for (int k0 = 0; k0 < C_; k0 += 4) {
            const int ka = k0 + 2 * hf;
            v2f a, bb;
            a.x  = cov_lds[(mt * 16 + ln) * C_ + ka];       // A = cov[M][K]
            a.y  = cov_lds[(mt * 16 + ln) * C_ + ka + 1];
            bb.x = qb[(size_t)ka * HW_ + p];                // B[K][N] = qn[K][p]
            bb.y = qb[(size_t)(ka + 1) * HW_ + p];
            acc = wmma_f32_4(a, bb, acc);
        }
        // diagonal: this lane holds T[mt*16 + 8*hf + i][p] in acc[i]
        const int mb = mt * 16 + 8 * hf;
#pragma unroll
        for (int i = 0; i < 8; ++i) {
            partial += qb[(size_t)(mb + i) * HW_ + p] * acc[i];
        }
    }

    // combine the two lane-halves (same pixel p, complementary M ranges)
    partial += __shfl_xor(partial, 16, 32);       // full sim[b,j,p] in every lane

    // LeakyReLU(0.2) then weight by conv_w[p]
    float x = (partial >= 0.f) ? partial : 0.2f * partial;
    float val = x * cw[p];

    // 32-lane reduce counts each of the wave's 16 pixels twice -> *0.5
#pragma unroll
    for (int off = 16; off >= 1; off >>= 1) {
        val += __shfl_xor(val, off, 32);
    }
    if (lane == 0) {
        atomicAdd(out + b * WAYS_ + j, 0.5f * val);
    }
}

// ---------------------------------------------------------------------------
// Launch
// d_in[0] = q       [64,128,32,32] f32
// d_in[1] = support [10,5,128,32,32] f32
// d_in[2] = conv_w  [1024] f32
// d_out   = [64,10] f32
// d_ws    : qn (B*C*HW f32 = 32 MB) | cov (WAYS*C*C f32 = 640 KB)
// ---------------------------------------------------------------------------
extern "C" void kernel_launch(void* const* d_in, const int* in_sizes, int n_in,
                              void* d_out, int out_size, void* d_ws, size_t ws_size,
                              hipStream_t stream) {
    const float* q   = (const float*)d_in[0];
    const float* sup = (const float*)d_in[1];
    const float* cw  = (const float*)d_in[2];
    float* out = (float*)d_out;

    float* qn  = (float*)d_ws;                       // B*C*HW floats
    float* cov = qn + (size_t)B_ * C_ * HW_;         // WAYS*C*C floats

    zero_kernel<<<dim3((B_ * WAYS_ + 255) / 256), dim3(256), 0, stream>>>(out, B_ * WAYS_);
    rownorm_kernel<<<dim3(B_ * C_), dim3(256), 0, stream>>>(q, qn);
    cov_kernel<<<dim3(WAYS_), dim3(256), 0, stream>>>(sup, cov);
    sim_kernel<<<dim3(HW_ / 128, WAYS_, B_), dim3(256), 65536, stream>>>(qn, cov, cw, out);
}